// S4Model_15384572854568
// MI455X (gfx1250) — compile-verified
//
#include <hip/hip_runtime.h>
#include <math.h>

#define B_  16
#define L_  16384
#define H_  128
#define N2_ 32
#define NL_ 4
#define NC_ 35

typedef float v2f __attribute__((ext_vector_type(2)));
typedef float v8f __attribute__((ext_vector_type(8)));

__device__ __forceinline__ float gelu_exact(float x) {
    return 0.5f * x * (1.0f + erff(x * 0.70710678118654752440f));
}

// ---------------- per-layer SSM parameter precompute ----------------
// w = exp(dt*A), Ceff = 2*(C_re+iC_im)*(w-1)/A   (factor 2 folded in)
__global__ void precompute_kernel(const float* __restrict__ log_dt,
                                  const float* __restrict__ log_A_real,
                                  const float* __restrict__ A_imag,
                                  const float* __restrict__ C_re,
                                  const float* __restrict__ C_im,
                                  float* __restrict__ wr, float* __restrict__ wi,
                                  float* __restrict__ cr, float* __restrict__ ci) {
    int idx = blockIdx.x * blockDim.x + threadIdx.x;
    if (idx >= NL_ * H_ * N2_) return;
    int lh = idx / N2_;                       // layer*H + h
    float dt = expf(log_dt[lh]);
    float Ar = -expf(log_A_real[idx]);
    float Ai = A_imag[idx];
    float er = expf(dt * Ar);
    float wre = er * cosf(dt * Ai);
    float wim = er * sinf(dt * Ai);
    float nr = wre - 1.0f, ni = wim;
    float inv = 1.0f / (Ar * Ar + Ai * Ai);
    float qr = (nr * Ar + ni * Ai) * inv;     // (w-1)/A
    float qi = (ni * Ar - nr * Ai) * inv;
    float Cr = C_re[idx], Ci = C_im[idx];
    wr[idx] = wre; wi[idx] = wim;
    cr[idx] = 2.0f * (Cr * qr - Ci * qi);
    ci[idx] = 2.0f * (Cr * qi + Ci * qr);
}

// ---------------- encoder: h[b,l,:] = x[b,l]*enc_W + enc_b ----------------
__global__ void encode_kernel(const float* __restrict__ x,
                              const float* __restrict__ encW,
                              const float* __restrict__ encb,
                              float* __restrict__ h) {
    size_t idx = (size_t)blockIdx.x * blockDim.x + threadIdx.x;
    int hid = (int)(idx % H_);
    size_t row = idx / H_;                    // b*L + l
    h[idx] = fmaf(x[row], encW[hid], encb[hid]);
}

// ---------------- prenorm LayerNorm over H, fused transpose to (B,H,L) ----------------
__global__ void __launch_bounds__(256) ln_transpose_kernel(const float* __restrict__ h,
                                                           const float* __restrict__ nw,
                                                           const float* __restrict__ nb,
                                                           float* __restrict__ u) {
    __shared__ float tile[32][H_ + 1];        // padded: conflict-free transpose
    int b = blockIdx.y;
    int l0 = blockIdx.x * 32;
    int wv = threadIdx.x >> 5, lane = threadIdx.x & 31;
    #pragma unroll
    for (int r = 0; r < 4; ++r) {
        int lrow = wv * 4 + r;
        const float* row = h + ((size_t)b * L_ + l0 + lrow) * H_;
        float x0 = row[lane], x1 = row[lane + 32], x2 = row[lane + 64], x3 = row[lane + 96];
        float s = x0 + x1 + x2 + x3;
        float q = x0 * x0 + x1 * x1 + x2 * x2 + x3 * x3;
        #pragma unroll
        for (int m = 16; m >= 1; m >>= 1) {
            s += __shfl_xor(s, m, 32);
            q += __shfl_xor(q, m, 32);
        }
        float mean = s * (1.0f / H_);
        float var  = q * (1.0f / H_) - mean * mean;
        float inv  = rsqrtf(var + 1e-5f);
        tile[lrow][lane]      = (x0 - mean) * inv * nw[lane]      + nb[lane];
        tile[lrow][lane + 32] = (x1 - mean) * inv * nw[lane + 32] + nb[lane + 32];
        tile[lrow][lane + 64] = (x2 - mean) * inv * nw[lane + 64] + nb[lane + 64];
        tile[lrow][lane + 96] = (x3 - mean) * inv * nw[lane + 96] + nb[lane + 96];
    }
    __syncthreads();
    #pragma unroll
    for (int k = 0; k < 16; ++k) {
        int hrow = wv + 8 * k;
        u[(size_t)b * H_ * L_ + (size_t)hrow * L_ + l0 + lane] = tile[lane][hrow];
    }
}

// ---------------- diagonal SSM scan: one wave per (b,h), one mode per lane ----------------
// In-place on u: reads LN'd input, writes gelu(y + D*u).
__global__ void __launch_bounds__(256) scan_kernel(float* __restrict__ u,
                                                   const float* __restrict__ wr_,
                                                   const float* __restrict__ wi_,
                                                   const float* __restrict__ cr_,
                                                   const float* __restrict__ ci_,
                                                   const float* __restrict__ Dp) {
    __shared__ float tile[8][32][33];
    int wv = threadIdx.x >> 5, lane = threadIdx.x & 31;
    int seq = blockIdx.x * 8 + wv;            // 0 .. B*H-1
    int b = seq / H_, hh = seq % H_;
    float wre = wr_[hh * N2_ + lane];
    float wim = wi_[hh * N2_ + lane];
    float cre = cr_[hh * N2_ + lane];
    float cim = ci_[hh * N2_ + lane];
    float Dv  = Dp[hh];
    float sre = 0.f, sim = 0.f;
    float* up = u + (size_t)b * H_ * L_ + (size_t)hh * L_;
    for (int l0 = 0; l0 < L_; l0 += 32) {
        float uval = up[l0 + lane];           // coalesced tile load
        #pragma unroll
        for (int t = 0; t < 32; ++t) {
            float ub = __shfl(uval, t, 32);   // broadcast u[l0+t]
            float nre = fmaf(wre, sre, fmaf(-wim, sim, ub));
            float nim = fmaf(wim, sre, wre * sim);
            sre = nre; sim = nim;
            tile[wv][t][lane] = fmaf(cre, sre, -cim * sim);  // Re(C*s)
        }
        __syncthreads();
        float y = 0.f;
        #pragma unroll
        for (int j = 0; j < 32; ++j) y += tile[wv][lane][j]; // reduce 32 modes
        float v = y + Dv * uval;              // skip connection
        up[l0 + lane] = gelu_exact(v);        // coalesced tile store (in-place safe)
        __syncthreads();
    }
}

// ---------------- WMMA f32 GEMM: z = W(2H,H) @ g(H,L), fused GLU + residual ----------------
// D(m=l, n=o) = A(g^T) x B(W^T).  Wave wv handles o-tile [16wv,16wv+16) and its gate twin.
__global__ void __launch_bounds__(256) gemm_glu_kernel(const float* __restrict__ g,
                                                       const float* __restrict__ W,
                                                       const float* __restrict__ bias,
                                                       float* __restrict__ h) {
    int wv = threadIdx.x >> 5, lane = threadIdx.x & 31;
    int lm = lane & 15, lh = lane >> 4;
    int o0 = wv * 16;
    int l0 = blockIdx.x * 16;
    int b  = blockIdx.y;
    const float* gb = g + (size_t)b * H_ * L_;
    v8f ca = {};
    v8f cg = {};
    for (int k0 = 0; k0 < H_; k0 += 4) {
        int kk = k0 + 2 * lh;
        v2f A;                                // A[m=l][k] = g[k][l0+m] (lane-contiguous in l)
        A.x = gb[(size_t)kk * L_ + l0 + lm];
        A.y = gb[(size_t)(kk + 1) * L_ + l0 + lm];
        v2f Ba;                               // B[k][n=o] = W[o0+n][k]
        Ba.x = W[(o0 + lm) * H_ + kk];
        Ba.y = W[(o0 + lm) * H_ + kk + 1];
        v2f Bg;                               // gate half: rows o0+128
        Bg.x = W[(H_ + o0 + lm) * H_ + kk];
        Bg.y = W[(H_ + o0 + lm) * H_ + kk + 1];
        ca = __builtin_amdgcn_wmma_f32_16x16x4_f32(false, A, false, Ba, (short)0, ca, false, false);
        cg = __builtin_amdgcn_wmma_f32_16x16x4_f32(false, A, false, Bg, (short)0, cg, false, false);
    }
    float bo_a = bias[o0 + lm];
    float bo_g = bias[H_ + o0 + lm];
    #pragma unroll
    for (int r = 0; r < 8; ++r) {
        int l = l0 + r + 8 * lh;              // C/D layout: M = r (+8 for upper lanes)
        size_t addr = ((size_t)b * L_ + l) * H_ + o0 + lm;
        float a  = ca[r] + bo_a;
        float gg = cg[r] + bo_g;
        h[addr] = fmaf(a, 1.0f / (1.0f + expf(-gg)), h[addr]);  // GLU + residual, in-place
    }
}

// ---------------- pooling + head ----------------
__global__ void zero_kernel(float* p, int n) {
    int i = blockIdx.x * blockDim.x + threadIdx.x;
    if (i < n) p[i] = 0.f;
}

__global__ void pool_kernel(const float* __restrict__ h, float* __restrict__ pooled) {
    int b  = blockIdx.y;
    int l0 = blockIdx.x * 256;
    int hh = threadIdx.x;                     // 0..127
    float s = 0.f;
    const float* base = h + ((size_t)b * L_ + l0) * H_ + hh;
    for (int l = 0; l < 256; ++l) s += base[(size_t)l * H_];
    atomicAdd(&pooled[b * H_ + hh], s);
}

__global__ void __launch_bounds__(128) head_kernel(const float* __restrict__ pooled,
                                                   const float* __restrict__ cnw,
                                                   const float* __restrict__ cnb,
                                                   const float* __restrict__ W1,
                                                   const float* __restrict__ b1,
                                                   const float* __restrict__ W2,
                                                   const float* __restrict__ b2,
                                                   float* __restrict__ out) {
    __shared__ float xs[H_];
    __shared__ float h1[H_];
    __shared__ float rs[4], rq[4];
    int b = blockIdx.x, tid = threadIdx.x;
    int wv = tid >> 5, lane = tid & 31;
    float v = pooled[b * H_ + tid] * (1.0f / L_);   // finish the mean
    float s = v, q = v * v;
    #pragma unroll
    for (int m = 16; m >= 1; m >>= 1) { s += __shfl_xor(s, m, 32); q += __shfl_xor(q, m, 32); }
    if (lane == 0) { rs[wv] = s; rq[wv] = q; }
    __syncthreads();
    float st = rs[0] + rs[1] + rs[2] + rs[3];
    float qt = rq[0] + rq[1] + rq[2] + rq[3];
    float mean = st * (1.0f / H_);
    float var  = qt * (1.0f / H_) - mean * mean;
    float inv  = rsqrtf(var + 1e-5f);
    xs[tid] = (v - mean) * inv * cnw[tid] + cnb[tid];
    __syncthreads();
    float acc = b1[tid];
    for (int j = 0; j < H_; ++j) acc = fmaf(xs[j], W1[j * H_ + tid], acc);
    h1[tid] = gelu_exact(acc);
    __syncthreads();
    if (tid < NC_) {
        float o = b2[tid];
        for (int j = 0; j < H_; ++j) o = fmaf(h1[j], W2[j * NC_ + tid], o);
        out[b * NC_ + tid] = o;
    }
}

extern "C" void kernel_launch(void* const* d_in, const int* in_sizes, int n_in,
                              void* d_out, int out_size, void* d_ws, size_t ws_size,
                              hipStream_t stream) {
    const float* x          = (const float*)d_in[0];
    const float* enc_W      = (const float*)d_in[1];
    const float* enc_b      = (const float*)d_in[2];
    const float* log_dt     = (const float*)d_in[3];
    const float* log_A_real = (const float*)d_in[4];
    const float* A_imag     = (const float*)d_in[5];
    const float* C_re       = (const float*)d_in[6];
    const float* C_im       = (const float*)d_in[7];
    const float* Dp         = (const float*)d_in[8];
    const float* out_W      = (const float*)d_in[9];
    const float* out_b      = (const float*)d_in[10];
    const float* norm_w     = (const float*)d_in[11];
    const float* norm_b     = (const float*)d_in[12];
    const float* cls_norm_w = (const float*)d_in[13];
    const float* cls_norm_b = (const float*)d_in[14];
    const float* cls_W1     = (const float*)d_in[15];
    const float* cls_b1     = (const float*)d_in[16];
    const float* cls_W2     = (const float*)d_in[17];
    const float* cls_b2     = (const float*)d_in[18];
    float* out = (float*)d_out;

    float* ws = (float*)d_ws;
    const size_t BLH = (size_t)B_ * L_ * H_;
    const int NLHN = NL_ * H_ * N2_;
    float* h_buf  = ws;                       // (B,L,H) residual stream
    float* u_buf  = ws + BLH;                 // (B,H,L) LN'd input -> gelu activations
    float* par    = u_buf + BLH;
    float* p_wr   = par;
    float* p_wi   = par + 1 * NLHN;
    float* p_cr   = par + 2 * NLHN;
    float* p_ci   = par + 3 * NLHN;
    float* pooled = par + 4 * NLHN;

    precompute_kernel<<<(NLHN + 255) / 256, 256, 0, stream>>>(
        log_dt, log_A_real, A_imag, C_re, C_im, p_wr, p_wi, p_cr, p_ci);

    encode_kernel<<<(int)(BLH / 256), 256, 0, stream>>>(x, enc_W, enc_b, h_buf);

    for (int i = 0; i < NL_; ++i) {
        dim3 gln(L_ / 32, B_);
        ln_transpose_kernel<<<gln, 256, 0, stream>>>(h_buf, norm_w + i * H_, norm_b + i * H_, u_buf);
        scan_kernel<<<(B_ * H_) / 8, 256, 0, stream>>>(
            u_buf, p_wr + i * H_ * N2_, p_wi + i * H_ * N2_,
            p_cr + i * H_ * N2_, p_ci + i * H_ * N2_, Dp + i * H_);
        dim3 gmm(L_ / 16, B_);
        gemm_glu_kernel<<<gmm, 256, 0, stream>>>(
            u_buf, out_W + (size_t)i * 2 * H_ * H_, out_b + i * 2 * H_, h_buf);
    }

    zero_kernel<<<(B_ * H_ + 255) / 256, 256, 0, stream>>>(pooled, B_ * H_);
    dim3 gp(L_ / 256, B_);
    pool_kernel<<<gp, 128, 0, stream>>>(h_buf, pooled);
    head_kernel<<<B_, 128, 0, stream>>>(pooled, cls_norm_w, cls_norm_b,
                                        cls_W1, cls_b1, cls_W2, cls_b2, out);

    (void)in_sizes; (void)n_in; (void)out_size; (void)ws_size;
}